// NeuralMemory_77326591197222
// MI455X (gfx1250) — compile-verified
//
#include <hip/hip_runtime.h>
#include <cstdint>

typedef float v2f __attribute__((ext_vector_type(2)));
typedef float v8f __attribute__((ext_vector_type(8)));

namespace {

constexpr int B_ = 2, T_ = 64, C_ = 384, H_ = 768;
constexpr float LR = 0.01f, MOM = 0.9f, DECAY = 0.001f;

// ---- workspace layout (float offsets) ----
constexpr int OFF_W    = 0;                     // w[0..63], ws[64] = (1-DECAY)^T
constexpr int OFF_K    = 128;                   // B*T*C
constexpr int OFF_V    = OFF_K    + B_*T_*C_;   // B*T*C
constexpr int OFF_H1   = OFF_V    + B_*T_*C_;   // B*T*H
constexpr int OFF_GP   = OFF_H1   + B_*T_*H_;   // B*T*H  (SiLU' factor)
constexpr int OFF_DP   = OFF_GP   + B_*T_*H_;   // B*T*C
constexpr int OFF_WDPT = OFF_DP   + B_*T_*C_;   // B*C*T  (w[t]*dp, transposed)
constexpr int OFF_WDHT = OFF_WDPT + B_*C_*T_;   // B*H*T  (w[t]*dh1p, transposed)

__device__ __forceinline__ v8f wmma4(v2f a, v2f b, v8f c) {
  // D = A(16x4 f32) * B(4x16 f32) + C(16x16 f32)
  return __builtin_amdgcn_wmma_f32_16x16x4_f32(false, a, false, b, (short)0, c,
                                               false, false);
}

// One wave computes a 32x32 output tile. A is MxK row-major (lda), B is KxN.
// BT=true : B[k][n] lives at Bm + n*ldb + k (transposed weight, contiguous in k)
// BT=false: B[k][n] lives at Bm + k*ldb + n (normal row-major)
template <bool BT>
__device__ __forceinline__ void gemm32x32(const float* __restrict__ A, int lda,
                                          const float* __restrict__ Bm, int ldb,
                                          int K, int m0, int n0, int lane,
                                          v8f acc[4]) {
  const int mr = lane & 15;
  const int kh = (lane >> 4) << 1;   // 0 for lanes 0-15, 2 for lanes 16-31
  for (int kk = 0; kk < K; kk += 4) {
    const int kp = kk + kh;
    v2f a0 = *reinterpret_cast<const v2f*>(A + (m0 + mr) * lda + kp);
    v2f a1 = *reinterpret_cast<const v2f*>(A + (m0 + 16 + mr) * lda + kp);
    v2f b0, b1;
    if (BT) {
      b0 = *reinterpret_cast<const v2f*>(Bm + (n0 + mr) * ldb + kp);
      b1 = *reinterpret_cast<const v2f*>(Bm + (n0 + 16 + mr) * ldb + kp);
    } else {
      const float* pb = Bm + kp * ldb + n0 + mr;
      b0.x = pb[0];  b0.y = pb[ldb];
      b1.x = pb[16]; b1.y = pb[ldb + 16];
    }
    acc[0] = wmma4(a0, b0, acc[0]);
    acc[1] = wmma4(a0, b1, acc[1]);
    acc[2] = wmma4(a1, b0, acc[2]);
    acc[3] = wmma4(a1, b1, acc[3]);
  }
}

// ---- kernel 0: scan weights (pscan collapsed to closed form) ----
__global__ void k_init_w(float* __restrict__ ws) {
  int t = threadIdx.x;  // 64 threads
  const float a = 1.0f - DECAY;
  float n = (float)(T_ - t);
  float an = __powf(a, n), mn = __powf(MOM, n);
  ws[OFF_W + t] = -LR * (1.0f - MOM) * (an - mn) / (a - MOM);
  if (t == 0) ws[OFF_W + 64] = __powf(a, (float)T_);
}

// ---- kernel 1: k = x@Wk^T + bk ; v = x@Wv^T + bv ----
__global__ void k_gemm_kv(const float* __restrict__ x,
                          const float* __restrict__ Wk, const float* __restrict__ bk,
                          const float* __restrict__ Wv, const float* __restrict__ bv,
                          float* __restrict__ ws) {
  const int b = blockIdx.y, which = blockIdx.z, lane = threadIdx.x;
  const int m0 = (blockIdx.x & 1) * 32, n0 = (blockIdx.x >> 1) * 32;
  const float* A = x + b * T_ * C_;
  const float* W = which ? Wv : Wk;
  const float* bias = which ? bv : bk;
  float* out = ws + (which ? OFF_V : OFF_K) + b * T_ * C_;
  v8f acc[4] = {};
  gemm32x32<true>(A, C_, W, C_, C_, m0, n0, lane, acc);
  const int cr = lane & 15, hi = lane >> 4;
#pragma unroll
  for (int mi = 0; mi < 2; ++mi)
#pragma unroll
    for (int ni = 0; ni < 2; ++ni) {
      int col = n0 + ni * 16 + cr;
      float bb = bias[col];
      v8f a = acc[mi * 2 + ni];
#pragma unroll
      for (int r = 0; r < 8; ++r) {
        int row = m0 + mi * 16 + r + 8 * hi;
        out[row * C_ + col] = a[r] + bb;
      }
    }
}

// ---- kernel 2: h1_pre = k@sW0^T + sb0 ; store SiLU(h1_pre) and SiLU' ----
__global__ void k_gemm_h1(const float* __restrict__ sW0,
                          const float* __restrict__ sb0,
                          float* __restrict__ ws) {
  const int b = blockIdx.y, lane = threadIdx.x;
  const int m0 = (blockIdx.x % 2) * 32, n0 = (blockIdx.x / 2) * 32;  // N=H: 24 tiles
  const float* A = ws + OFF_K + b * T_ * C_;
  const float* W = sW0 + b * H_ * C_;
  float* h1 = ws + OFF_H1 + b * T_ * H_;
  float* gp = ws + OFF_GP + b * T_ * H_;
  v8f acc[4] = {};
  gemm32x32<true>(A, C_, W, C_, C_, m0, n0, lane, acc);
  const int cr = lane & 15, hi = lane >> 4;
#pragma unroll
  for (int mi = 0; mi < 2; ++mi)
#pragma unroll
    for (int ni = 0; ni < 2; ++ni) {
      int col = n0 + ni * 16 + cr;
      float bb = sb0[b * H_ + col];
      v8f a = acc[mi * 2 + ni];
#pragma unroll
      for (int r = 0; r < 8; ++r) {
        int row = m0 + mi * 16 + r + 8 * hi;
        float pre = a[r] + bb;
        float sig = 1.0f / (1.0f + __expf(-pre));
        h1[row * H_ + col] = pre * sig;
        gp[row * H_ + col] = sig * (1.0f + pre * (1.0f - sig));
      }
    }
}

// ---- kernel 3: p = h1@sW1^T + sb1 ; dp = p - v ; wdpT[c,t] = w[t]*dp[t,c] ----
__global__ void k_gemm_dp(const float* __restrict__ sW1,
                          const float* __restrict__ sb1,
                          float* __restrict__ ws) {
  const int b = blockIdx.y, lane = threadIdx.x;
  const int m0 = (blockIdx.x % 2) * 32, n0 = (blockIdx.x / 2) * 32;  // N=C: 12 tiles
  const float* A = ws + OFF_H1 + b * T_ * H_;
  const float* W = sW1 + b * C_ * H_;
  const float* v = ws + OFF_V + b * T_ * C_;
  float* dp = ws + OFF_DP + b * T_ * C_;
  float* wdpT = ws + OFF_WDPT + b * C_ * T_;
  v8f acc[4] = {};
  gemm32x32<true>(A, H_, W, H_, H_, m0, n0, lane, acc);
  const int cr = lane & 15, hi = lane >> 4;
#pragma unroll
  for (int mi = 0; mi < 2; ++mi)
#pragma unroll
    for (int ni = 0; ni < 2; ++ni) {
      int col = n0 + ni * 16 + cr;
      float bb = sb1[b * C_ + col];
      v8f a = acc[mi * 2 + ni];
#pragma unroll
      for (int r = 0; r < 8; ++r) {
        int row = m0 + mi * 16 + r + 8 * hi;
        float d = (a[r] + bb) - v[row * C_ + col];
        dp[row * C_ + col] = d;
        wdpT[col * T_ + row] = ws[OFF_W + row] * d;
      }
    }
}

// ---- kernel 4: dh1 = dp@sW1 (normal B) ; wdh1pT[h,t] = w[t]*dh1*SiLU' ----
__global__ void k_gemm_dh1(const float* __restrict__ sW1,
                           float* __restrict__ ws) {
  const int b = blockIdx.y, lane = threadIdx.x;
  const int m0 = (blockIdx.x % 2) * 32, n0 = (blockIdx.x / 2) * 32;  // N=H: 24 tiles
  const float* A = ws + OFF_DP + b * T_ * C_;
  const float* W = sW1 + b * C_ * H_;
  const float* gp = ws + OFF_GP + b * T_ * H_;
  float* wdhT = ws + OFF_WDHT + b * H_ * T_;
  v8f acc[4] = {};
  gemm32x32<false>(A, C_, W, H_, C_, m0, n0, lane, acc);
  const int cr = lane & 15, hi = lane >> 4;
#pragma unroll
  for (int mi = 0; mi < 2; ++mi)
#pragma unroll
    for (int ni = 0; ni < 2; ++ni) {
      int col = n0 + ni * 16 + cr;
      v8f a = acc[mi * 2 + ni];
#pragma unroll
      for (int r = 0; r < 8; ++r) {
        int row = m0 + mi * 16 + r + 8 * hi;
        float d = a[r] * gp[row * H_ + col];
        wdhT[col * T_ + row] = ws[OFF_W + row] * d;
      }
    }
}

// ---- kernel 5: nW0 = a^T*sW0 + wdh1pT @ k   (HxT @ TxC) ----
__global__ void k_gemm_nw0(const float* __restrict__ sW0,
                           float* __restrict__ nW0,
                           float* __restrict__ ws) {
  const int b = blockIdx.y, lane = threadIdx.x;
  const int m0 = (blockIdx.x / 12) * 32, n0 = (blockIdx.x % 12) * 32;  // 24 x 12 tiles
  const float* A = ws + OFF_WDHT + b * H_ * T_;
  const float* Bm = ws + OFF_K + b * T_ * C_;
  const float dT = ws[OFF_W + 64];
  float* out = nW0 + b * H_ * C_;
  const float* w0 = sW0 + b * H_ * C_;
  v8f acc[4] = {};
  gemm32x32<false>(A, T_, Bm, C_, T_, m0, n0, lane, acc);
  const int cr = lane & 15, hi = lane >> 4;
#pragma unroll
  for (int mi = 0; mi < 2; ++mi)
#pragma unroll
    for (int ni = 0; ni < 2; ++ni) {
      int col = n0 + ni * 16 + cr;
      v8f a = acc[mi * 2 + ni];
#pragma unroll
      for (int r = 0; r < 8; ++r) {
        int row = m0 + mi * 16 + r + 8 * hi;
        out[row * C_ + col] = dT * w0[row * C_ + col] + a[r];
      }
    }
}

// ---- kernel 6: nW1 = a^T*sW1 + wdpT @ h1   (CxT @ TxH) ----
__global__ void k_gemm_nw1(const float* __restrict__ sW1,
                           float* __restrict__ nW1,
                           float* __restrict__ ws) {
  const int b = blockIdx.y, lane = threadIdx.x;
  const int m0 = (blockIdx.x / 24) * 32, n0 = (blockIdx.x % 24) * 32;  // 12 x 24 tiles
  const float* A = ws + OFF_WDPT + b * C_ * T_;
  const float* Bm = ws + OFF_H1 + b * T_ * H_;
  const float dT = ws[OFF_W + 64];
  float* out = nW1 + b * C_ * H_;
  const float* w1 = sW1 + b * C_ * H_;
  v8f acc[4] = {};
  gemm32x32<false>(A, T_, Bm, H_, T_, m0, n0, lane, acc);
  const int cr = lane & 15, hi = lane >> 4;
#pragma unroll
  for (int mi = 0; mi < 2; ++mi)
#pragma unroll
    for (int ni = 0; ni < 2; ++ni) {
      int col = n0 + ni * 16 + cr;
      v8f a = acc[mi * 2 + ni];
#pragma unroll
      for (int r = 0; r < 8; ++r) {
        int row = m0 + mi * 16 + r + 8 * hi;
        out[row * H_ + col] = dT * w1[row * H_ + col] + a[r];
      }
    }
}

// ---- kernel 7: bias updates (row-sums of the transposed weighted grads) ----
__global__ void k_bias(const float* __restrict__ sb0, const float* __restrict__ sb1,
                       float* __restrict__ nb0, float* __restrict__ nb1,
                       const float* __restrict__ ws) {
  int idx = blockIdx.x * blockDim.x + threadIdx.x;
  if (idx >= B_ * (H_ + C_)) return;
  int b = idx / (H_ + C_), r = idx % (H_ + C_);
  const float dT = ws[OFF_W + 64];
  if (r < H_) {
    const float* p = ws + OFF_WDHT + b * H_ * T_ + r * T_;
    float s = 0.f;
#pragma unroll 4
    for (int t = 0; t < T_; ++t) s += p[t];
    nb0[b * H_ + r] = dT * sb0[b * H_ + r] + s;
  } else {
    int c = r - H_;
    const float* p = ws + OFF_WDPT + b * C_ * T_ + c * T_;
    float s = 0.f;
#pragma unroll 4
    for (int t = 0; t < T_; ++t) s += p[t];
    nb1[b * C_ + c] = dT * sb1[b * C_ + c] + s;
  }
}

// ---- kernel 8: retrieval (4 chained matvecs through LDS) ----
__global__ void k_retrieve(const float* __restrict__ Wq, const float* __restrict__ bq,
                           const float* __restrict__ Wo, const float* __restrict__ bo,
                           const float* __restrict__ iq,
                           const float* __restrict__ sW0, const float* __restrict__ sb0,
                           const float* __restrict__ sW1, const float* __restrict__ sb1,
                           float* __restrict__ out) {
  __shared__ float q[C_], h[H_], pred[C_];
  const int b = blockIdx.x, t = threadIdx.x;  // 384 threads
  // q = Wq @ iq + bq
  {
    const float* wr = Wq + t * C_;
    float s = 0.f;
    for (int c = 0; c < C_; ++c) s += wr[c] * iq[c];
    q[t] = s + bq[t];
  }
  __syncthreads();
  // h = SiLU(sW0 @ q + sb0)
#pragma unroll
  for (int j = 0; j < 2; ++j) {
    int hh = t + j * C_;
    const float* w0 = sW0 + b * H_ * C_ + hh * C_;
    float s = 0.f;
    for (int c = 0; c < C_; ++c) s += w0[c] * q[c];
    s += sb0[b * H_ + hh];
    float sig = 1.0f / (1.0f + __expf(-s));
    h[hh] = s * sig;
  }
  __syncthreads();
  // pred = sW1 @ h + sb1
  {
    const float* w1 = sW1 + b * C_ * H_ + t * H_;
    float s = 0.f;
    for (int hh = 0; hh < H_; ++hh) s += w1[hh] * h[hh];
    pred[t] = s + sb1[b * C_ + t];
  }
  __syncthreads();
  // out = Wo @ pred + bo
  {
    const float* wo = Wo + t * C_;
    float s = 0.f;
    for (int c = 0; c < C_; ++c) s += wo[c] * pred[c];
    out[b * C_ + t] = s + bo[t];
  }
}

}  // namespace

extern "C" void kernel_launch(void* const* d_in, const int* in_sizes, int n_in,
                              void* d_out, int out_size, void* d_ws, size_t ws_size,
                              hipStream_t stream) {
  const float* x   = (const float*)d_in[0];
  const float* Wq  = (const float*)d_in[1];
  const float* bq  = (const float*)d_in[2];
  const float* Wk  = (const float*)d_in[3];
  const float* bk  = (const float*)d_in[4];
  const float* Wv  = (const float*)d_in[5];
  const float* bv  = (const float*)d_in[6];
  const float* Wo  = (const float*)d_in[7];
  const float* bo  = (const float*)d_in[8];
  const float* iq  = (const float*)d_in[9];
  const float* sW0 = (const float*)d_in[10];
  const float* sb0 = (const float*)d_in[11];
  const float* sW1 = (const float*)d_in[12];
  const float* sb1 = (const float*)d_in[13];

  float* out = (float*)d_out;
  float* ws  = (float*)d_ws;

  float* out_out = out;                               // [B,1,C]
  float* out_nW0 = out + B_ * C_;                     // [B,H,C]
  float* out_nb0 = out_nW0 + B_ * H_ * C_;            // [B,H]
  float* out_nW1 = out_nb0 + B_ * H_;                 // [B,C,H]
  float* out_nb1 = out_nW1 + B_ * C_ * H_;            // [B,C]

  k_init_w <<<1, 64, 0, stream>>>(ws);
  k_gemm_kv<<<dim3(24, B_, 2), 32, 0, stream>>>(x, Wk, bk, Wv, bv, ws);
  k_gemm_h1<<<dim3(48, B_), 32, 0, stream>>>(sW0, sb0, ws);
  k_gemm_dp<<<dim3(24, B_), 32, 0, stream>>>(sW1, sb1, ws);
  k_gemm_dh1<<<dim3(48, B_), 32, 0, stream>>>(sW1, ws);
  k_gemm_nw0<<<dim3(288, B_), 32, 0, stream>>>(sW0, out_nW0, ws);
  k_gemm_nw1<<<dim3(288, B_), 32, 0, stream>>>(sW1, out_nW1, ws);
  k_bias    <<<dim3(18), 128, 0, stream>>>(sb0, sb1, out_nb0, out_nb1, ws);
  k_retrieve<<<dim3(B_), C_, 0, stream>>>(Wq, bq, Wo, bo, iq, sW0, sb0, sW1, sb1, out_out);
}